// AttentiveGraphConvolution_26637387170115
// MI455X (gfx1250) — compile-verified
//
#include <hip/hip_runtime.h>
#include <math.h>

#define N_NODES 50000
#define N_EDGES 800000
#define IN_CH   128
#define HEADS   4
#define CPH     16
#define HC      64          // HEADS*CPH
#define NEG_SLOPE 0.2f

typedef __attribute__((ext_vector_type(2))) float v2f;
typedef __attribute__((ext_vector_type(8))) float v8f;

// Order-preserving float<->int encoding so atomicMax(int) == max(float).
__device__ __forceinline__ int f2ord(float f) {
    int i = __float_as_int(f);
    return (i >= 0) ? i : (i ^ 0x7FFFFFFF);
}
__device__ __forceinline__ float ord2f(int i) {
    return __int_as_float((i >= 0) ? i : (i ^ 0x7FFFFFFF));
}
__device__ __forceinline__ float lrelu(float v) {
    return v > 0.0f ? v : NEG_SLOPE * v;
}

// -------- Kernel 1: h = x @ W via V_WMMA_F32_16X16X4_F32 ------------------
// One wave per 16x16 output tile. 3125 row tiles x 4 col tiles.
__global__ void gat_gemm_wmma(const float* __restrict__ x,
                              const float* __restrict__ W,
                              float* __restrict__ hbuf) {
    const int wave = (blockIdx.x * blockDim.x + threadIdx.x) >> 5;
    const int lane = threadIdx.x & 31;
    const int nColTiles = HC / 16;                       // 4
    const int nTiles = (N_NODES / 16) * nColTiles;       // 12500
    if (wave >= nTiles) return;                          // wave-uniform exit
    const int rowTile = wave / nColTiles;
    const int colTile = wave % nColTiles;
    const int half = lane >> 4;                          // 0/1
    const int l16  = lane & 15;
    const int row  = rowTile * 16 + l16;                 // A-matrix row (M)
    const int col  = colTile * 16 + l16;                 // B/C column (N)

    const float* __restrict__ xr = x + (size_t)row * IN_CH;
    v8f acc = {};
    #pragma unroll
    for (int k = 0; k < IN_CH; k += 4) {
        const int kk = k + 2 * half;                     // lane-half K pair
        v2f a, b;
        a.x = xr[kk];
        a.y = xr[kk + 1];
        b.x = W[(size_t)kk * HC + col];
        b.y = W[(size_t)(kk + 1) * HC + col];
        acc = __builtin_amdgcn_wmma_f32_16x16x4_f32(
            /*neg_a=*/false, a, /*neg_b=*/false, b,
            /*c_mod=*/(short)0, acc, /*reuse_a=*/false, /*reuse_b=*/false);
    }
    float* __restrict__ hr = hbuf + (size_t)(rowTile * 16) * HC + colTile * 16 + l16;
    #pragma unroll
    for (int j = 0; j < 8; ++j)
        hr[(size_t)(j + 8 * half) * HC] = acc[j];        // C/D: VGPR j -> M = j + 8*half
}

// -------- Kernel 2: per-node attention scalars + state init ---------------
// N*64 threads: zero out[]; first 4 threads per node also compute a_src/a_dst,
// init m (ordered -inf) and denom (0).
__global__ void gat_node_prep(const float* __restrict__ hbuf,
                              const float* __restrict__ att_src,
                              const float* __restrict__ att_dst,
                              float* __restrict__ a_src, float* __restrict__ a_dst,
                              int* __restrict__ m_ord, float* __restrict__ denom,
                              float* __restrict__ out) {
    const int t = blockIdx.x * blockDim.x + threadIdx.x;
    if (t >= N_NODES * HC) return;
    const int n  = t >> 6;
    const int ch = t & 63;
    out[t] = 0.0f;
    if (ch < HEADS) {
        const int head = ch;
        const float* __restrict__ hr = hbuf + (size_t)n * HC + head * CPH;
        float ssum = 0.0f, dsum = 0.0f;
        #pragma unroll
        for (int c = 0; c < CPH; ++c) {
            const float hv = hr[c];
            ssum += hv * att_src[head * CPH + c];
            dsum += hv * att_dst[head * CPH + c];
        }
        a_src[n * HEADS + head] = ssum;
        a_dst[n * HEADS + head] = dsum;
        m_ord[n * HEADS + head] = f2ord(-INFINITY);
        denom[n * HEADS + head] = 0.0f;
    }
}

// -------- Kernel 3: edge pass A — segment max per (dst, head) -------------
__global__ void gat_edge_max(const int* __restrict__ ei,
                             const float* __restrict__ a_src,
                             const float* __restrict__ a_dst,
                             int* __restrict__ m_ord, int Etot) {
    const int t = blockIdx.x * blockDim.x + threadIdx.x;
    if (t >= Etot * HEADS) return;
    const int e = t >> 2;
    const int head = t & 3;
    int s, d;
    if (e < N_EDGES) { s = ei[e]; d = ei[N_EDGES + e]; }
    else             { s = e - N_EDGES; d = s; }          // self-loop
    const float v = lrelu(a_src[s * HEADS + head] + a_dst[d * HEADS + head]);
    atomicMax(&m_ord[d * HEADS + head], f2ord(v));
}

// -------- Kernel 4: edge pass B — fused exp / denom / weighted scatter ----
// One wave per edge; lane owns channels lane and lane+32 (head = ch>>4).
__global__ void gat_edge_accum(const int* __restrict__ ei,
                               const float* __restrict__ a_src,
                               const float* __restrict__ a_dst,
                               const int* __restrict__ m_ord,
                               const float* __restrict__ hbuf,
                               float* __restrict__ denom,
                               float* __restrict__ out, int Etot) {
    const int wave = (blockIdx.x * blockDim.x + threadIdx.x) >> 5;
    const int lane = threadIdx.x & 31;
    if (wave >= Etot) return;
    int s, d;
    if (wave < N_EDGES) { s = ei[wave]; d = ei[N_EDGES + wave]; }
    else                { s = wave - N_EDGES; d = s; }
    const int h0 = lane >> 4;       // heads 0/1 for channels 0..31
    const int h1 = h0 + 2;          // heads 2/3 for channels 32..63

    float e0 = lrelu(a_src[s * HEADS + h0] + a_dst[d * HEADS + h0]);
    float p0 = __expf(e0 - ord2f(m_ord[d * HEADS + h0]));
    float e1 = lrelu(a_src[s * HEADS + h1] + a_dst[d * HEADS + h1]);
    float p1 = __expf(e1 - ord2f(m_ord[d * HEADS + h1]));

    if ((lane & 15) == 0) {         // lanes 0,16 cover all 4 heads
        atomicAdd(&denom[d * HEADS + h0], p0);
        atomicAdd(&denom[d * HEADS + h1], p1);
    }
    const int ch0 = lane;
    const int ch1 = lane + 32;
    atomicAdd(&out[(size_t)d * HC + ch0], p0 * hbuf[(size_t)s * HC + ch0]);
    atomicAdd(&out[(size_t)d * HC + ch1], p1 * hbuf[(size_t)s * HC + ch1]);
}

// -------- Kernel 5: finalize — divide by denom, add bias ------------------
__global__ void gat_finalize(float* __restrict__ out,
                             const float* __restrict__ denom,
                             const float* __restrict__ bias) {
    const int t = blockIdx.x * blockDim.x + threadIdx.x;
    if (t >= N_NODES * HC) return;
    const int n = t >> 6;
    const int ch = t & 63;
    const int head = ch >> 4;
    out[t] = out[t] / (denom[n * HEADS + head] + 1e-16f) + bias[ch];
}

extern "C" void kernel_launch(void* const* d_in, const int* in_sizes, int n_in,
                              void* d_out, int out_size, void* d_ws, size_t ws_size,
                              hipStream_t stream) {
    const float* x       = (const float*)d_in[0];
    const int*   ei      = (const int*)  d_in[1];
    const float* W       = (const float*)d_in[2];
    const float* att_src = (const float*)d_in[3];
    const float* att_dst = (const float*)d_in[4];
    const float* bias    = (const float*)d_in[5];
    float* out = (float*)d_out;

    // Workspace layout (16 MB total)
    float* hbuf  = (float*)d_ws;                              // [N,64]
    float* asrc  = hbuf + (size_t)N_NODES * HC;               // [N,4]
    float* adst  = asrc + (size_t)N_NODES * HEADS;            // [N,4]
    int*   m_ord = (int*)(adst + (size_t)N_NODES * HEADS);    // [N,4]
    float* denom = (float*)(m_ord + (size_t)N_NODES * HEADS); // [N,4]

    const int Etot = N_EDGES + N_NODES;

    // 1) GEMM: 12500 tiles, 8 waves (tiles) per 256-thread block
    const int nTiles = (N_NODES / 16) * (HC / 16);
    gat_gemm_wmma<<<(nTiles + 7) / 8, 256, 0, stream>>>(x, W, hbuf);

    // 2) node prep (+ zero out)
    gat_node_prep<<<(N_NODES * HC + 255) / 256, 256, 0, stream>>>(
        hbuf, att_src, att_dst, asrc, adst, m_ord, denom, out);

    // 3) segment max over edges
    gat_edge_max<<<(Etot * HEADS + 255) / 256, 256, 0, stream>>>(
        ei, asrc, adst, m_ord, Etot);

    // 4) fused exp/denom/scatter (one wave per edge, 8 edges per block)
    gat_edge_accum<<<(Etot + 7) / 8, 256, 0, stream>>>(
        ei, asrc, adst, m_ord, hbuf, denom, out, Etot);

    // 5) finalize
    gat_finalize<<<(N_NODES * HC + 255) / 256, 256, 0, stream>>>(out, denom, bias);
}